// SymptomGraphModule_72249939853605
// MI455X (gfx1250) — compile-verified
//
#include <hip/hip_runtime.h>
#include <math.h>

#define N_NODES 50000
#define N_EDGE0 500000
#define E_TOT   (N_EDGE0 + N_NODES)   /* edges + self-loops */
#define NEG_SLOPE 0.2f
#define EPS_F 1e-16f
#define NEG_KEY 0x807FFFFF            /* ordered-int encoding of -inf */

typedef float v2f __attribute__((ext_vector_type(2)));
typedef float v8f __attribute__((ext_vector_type(8)));
typedef int   v4i __attribute__((ext_vector_type(4)));

// ---------- helpers ----------
__device__ __forceinline__ int f2ord(float f) {
    int i = __float_as_int(f);
    return (i >= 0) ? i : (i ^ 0x7FFFFFFF);
}
__device__ __forceinline__ float ord2f(int i) {
    return __int_as_float((i >= 0) ? i : (i ^ 0x7FFFFFFF));
}
__device__ __forceinline__ void edge_sd(const int* __restrict__ ei, int e, int& s, int& d) {
    if (e < N_EDGE0) { s = ei[e]; d = ei[N_EDGE0 + e]; }
    else             { s = e - N_EDGE0; d = s; }
}

// ---------- async global->LDS copy of 16B (guarded; fallback = sync copy) ----------
__device__ __forceinline__ void async_copy_f4(float* lds_dst, const float* gsrc) {
#if __has_builtin(__builtin_amdgcn_global_load_async_to_lds_b128)
    v4i* g  = (v4i*)gsrc;      // strip const + reinterpret (generic AS)
    v4i* l4 = (v4i*)lds_dst;
    __builtin_amdgcn_global_load_async_to_lds_b128(
        (__attribute__((address_space(1))) v4i*)g,
        (__attribute__((address_space(3))) v4i*)l4,
        0, 0);
#else
    float4 v = *(const float4*)gsrc;
    *(float4*)lds_dst = v;
#endif
}
__device__ __forceinline__ void async_join_barrier() {
#if __has_builtin(__builtin_amdgcn_global_load_async_to_lds_b128)
#  if __has_builtin(__builtin_amdgcn_s_wait_asynccnt)
    __builtin_amdgcn_s_wait_asynccnt(0);
#  else
    asm volatile("s_wait_asynccnt 0x0" ::: "memory");
#  endif
#endif
    __syncthreads();
}

// ---------- fill kernels (deterministic init each launch) ----------
__global__ void fill_f32(float* __restrict__ p, float v, int n) {
    int i = blockIdx.x * blockDim.x + threadIdx.x;
    if (i < n) p[i] = v;
}
__global__ void fill_i32(int* __restrict__ p, int v, int n) {
    int i = blockIdx.x * blockDim.x + threadIdx.x;
    if (i < n) p[i] = v;
}

// ---------- FP32 WMMA GEMM, LDS-staged B panel ----------
// Block = 256 threads (8 waves). Block computes up to 128 rows x 64 cols of C.
// B panel [KC x 64] is staged in LDS (async) and shared by all 8 waves; each wave
// holds one A fragment per k-step and issues 4 WMMAs (16x64 strip, 4 accumulators).
// Requires: 16 | M, 64 | N, 64 | K.
#define KC 64
#define BN 64
__global__ void gemm_wmma_tiled(const float* __restrict__ A, const float* __restrict__ B,
                                float* __restrict__ C, int M, int N, int K) {
    __shared__ float Bs[KC * BN];               // 16 KB
    const int tid  = threadIdx.x;
    const int wv   = tid >> 5;
    const int l    = tid & 31;
    const int half = l >> 4;                    // 0: lanes 0-15, 1: lanes 16-31
    const int mr   = l & 15;
    const int mtiles   = M >> 4;
    const int ntiles64 = N / BN;
    const int bm = blockIdx.x / ntiles64;
    const int bn = blockIdx.x % ntiles64;
    const int tm = bm * 8 + wv;
    const bool active = (tm < mtiles);          // wave-uniform predicate

    const float* Arow = A + (size_t)((active ? tm : 0) * 16 + mr) * K + 2 * half;

    v8f acc0 = {}, acc1 = {}, acc2 = {}, acc3 = {};

    for (int kb = 0; kb < K; kb += KC) {
        // cooperative async load: B[kb..kb+KC)[bn*64..+64) -> Bs, 4 float4 per thread
        int t4 = tid;
        #pragma unroll
        for (int it = 0; it < 4; ++it, t4 += 256) {
            int row = t4 >> 4;                  // 16 float4 per 64-col row
            int c4  = t4 & 15;
            async_copy_f4(&Bs[row * BN + c4 * 4],
                          B + (size_t)(kb + row) * N + bn * BN + c4 * 4);
        }
        async_join_barrier();

        if (active) {
            for (int k0 = 0; k0 < KC; k0 += 4) {
                v2f a; a.x = Arow[kb + k0]; a.y = Arow[kb + k0 + 1];
                const int r0 = (k0 + 2 * half) * BN;
                const int r1 = r0 + BN;
                v2f b0; b0.x = Bs[r0 +  0 + mr]; b0.y = Bs[r1 +  0 + mr];
                v2f b1; b1.x = Bs[r0 + 16 + mr]; b1.y = Bs[r1 + 16 + mr];
                v2f b2; b2.x = Bs[r0 + 32 + mr]; b2.y = Bs[r1 + 32 + mr];
                v2f b3; b3.x = Bs[r0 + 48 + mr]; b3.y = Bs[r1 + 48 + mr];
                acc0 = __builtin_amdgcn_wmma_f32_16x16x4_f32(false, a, false, b0, (short)0, acc0, false, false);
                acc1 = __builtin_amdgcn_wmma_f32_16x16x4_f32(false, a, false, b1, (short)0, acc1, false, false);
                acc2 = __builtin_amdgcn_wmma_f32_16x16x4_f32(false, a, false, b2, (short)0, acc2, false, false);
                acc3 = __builtin_amdgcn_wmma_f32_16x16x4_f32(false, a, false, b3, (short)0, acc3, false, false);
            }
        }
        __syncthreads();
    }

    if (active) {
        float* Crow = C + (size_t)(tm * 16 + 8 * half) * N + bn * BN + mr;
        #pragma unroll
        for (int i = 0; i < 8; ++i) {
            Crow[(size_t)i * N +  0] = acc0[i];
            Crow[(size_t)i * N + 16] = acc1[i];
            Crow[(size_t)i * N + 32] = acc2[i];
            Crow[(size_t)i * N + 48] = acc3[i];
        }
    }
}

// ---------- per-(node,head) attention coefficients: one wave per (n,h) ----------
__global__ void node_alpha(const float* __restrict__ h, const float* __restrict__ a_s,
                           const float* __restrict__ a_d, float* __restrict__ as_out,
                           float* __restrict__ ad_out, int H, int C) {
    int w = blockIdx.x * (blockDim.x >> 5) + (threadIdx.x >> 5);
    if (w >= N_NODES * H) return;
    int l = threadIdx.x & 31;
    int n = w / H, hh = w % H;
    const float* hv = h + (size_t)n * H * C + (size_t)hh * C;
    const float* av = a_s + hh * C;
    const float* dv = a_d + hh * C;
    float ss = 0.f, sd = 0.f;
    for (int c = l; c < C; c += 32) { float x = hv[c]; ss += x * av[c]; sd += x * dv[c]; }
    #pragma unroll
    for (int off = 16; off > 0; off >>= 1) {
        ss += __shfl_xor(ss, off, 32);
        sd += __shfl_xor(sd, off, 32);
    }
    if (l == 0) { as_out[w] = ss; ad_out[w] = sd; }
}

// ---------- edge logits + segment max (atomic on ordered-int) ----------
__global__ void edge_logits(const int* __restrict__ ei, const float* __restrict__ as,
                            const float* __restrict__ ad, float* __restrict__ ebuf,
                            int* __restrict__ mkey, int H) {
    int e = blockIdx.x * blockDim.x + threadIdx.x;
    if (e >= E_TOT) return;
    int s, d; edge_sd(ei, e, s, d);
    for (int hh = 0; hh < H; ++hh) {
        float v = as[s * H + hh] + ad[d * H + hh];
        v = (v > 0.f) ? v : NEG_SLOPE * v;
        ebuf[(size_t)e * H + hh] = v;
        atomicMax(&mkey[d * H + hh], f2ord(v));
    }
}

// ---------- segment softmax denominator ----------
__global__ void edge_expsum(const int* __restrict__ ei, const float* __restrict__ ebuf,
                            const int* __restrict__ mkey, float* __restrict__ den, int H) {
    int e = blockIdx.x * blockDim.x + threadIdx.x;
    if (e >= E_TOT) return;
    int s, d; edge_sd(ei, e, s, d);
    for (int hh = 0; hh < H; ++hh) {
        float ex = __expf(ebuf[(size_t)e * H + hh] - ord2f(mkey[d * H + hh]));
        atomicAdd(&den[d * H + hh], ex);
    }
}

// ---------- weighted gather-scatter: one wave per edge, lanes across channels ----------
__global__ void edge_scatter(const int* __restrict__ ei, const float* __restrict__ h,
                             const float* __restrict__ ebuf, const int* __restrict__ mkey,
                             const float* __restrict__ den, float* __restrict__ out,
                             int H, int C) {
    int e = blockIdx.x * (blockDim.x >> 5) + (threadIdx.x >> 5);
    if (e >= E_TOT) return;
    int l = threadIdx.x & 31;
    int s, d; edge_sd(ei, e, s, d);
    int HC = H * C;
    const float* hrow = h + (size_t)s * HC;
    float* orow = out + (size_t)d * HC;
    __builtin_prefetch(hrow + l * 4, 0, 1);        // global_prefetch_b8: pull gather row
    for (int c = l; c < HC; c += 32) {
        int hh = c / C;
        float m  = ord2f(mkey[d * H + hh]);
        float al = __expf(ebuf[(size_t)e * H + hh] - m) / (den[d * H + hh] + EPS_F);
        atomicAdd(&orow[c], hrow[c] * al);
    }
}

// ---------- bias + ELU (in place) ----------
__global__ void bias_elu(float* __restrict__ x, const float* __restrict__ b, int total, int F) {
    int i = blockIdx.x * blockDim.x + threadIdx.x;
    if (i >= total) return;
    float v = x[i] + b[i % F];
    x[i] = (v > 0.f) ? v : (__expf(v) - 1.f);
}

// ---------- column mean + bias, broadcast to batch rows ----------
__global__ void col_mean(const float* __restrict__ x, const float* __restrict__ b,
                         float* __restrict__ out, int F, int Brows) {
    __shared__ float red[256];
    int j = blockIdx.x;                       // one block per output column
    float s = 0.f;
    for (int n = threadIdx.x; n < N_NODES; n += blockDim.x) s += x[(size_t)n * F + j];
    red[threadIdx.x] = s;
    __syncthreads();
    for (int off = blockDim.x >> 1; off > 0; off >>= 1) {
        if (threadIdx.x < off) red[threadIdx.x] += red[threadIdx.x + off];
        __syncthreads();
    }
    if (threadIdx.x == 0) {
        float v = red[0] / (float)N_NODES + b[j];
        for (int r = 0; r < Brows; ++r) out[r * F + j] = v;
    }
}

extern "C" void kernel_launch(void* const* d_in, const int* in_sizes, int n_in,
                              void* d_out, int out_size, void* d_ws, size_t ws_size,
                              hipStream_t stream) {
    const int*   ei    = (const int*)  d_in[0];
    const float* xin   = (const float*)d_in[2];   // [50000,64]
    const float* W1    = (const float*)d_in[3];   // [64,512]
    const float* aS1   = (const float*)d_in[4];   // [4,128]
    const float* aD1   = (const float*)d_in[5];
    const float* b1    = (const float*)d_in[6];   // [512]
    const float* W2    = (const float*)d_in[7];   // [512,256]
    const float* aS2   = (const float*)d_in[8];   // [1,256]
    const float* aD2   = (const float*)d_in[9];
    const float* b2    = (const float*)d_in[10];  // [256]
    float* out = (float*)d_out;                   // [8,256]

    // ---- workspace layout (floats), reused between layers ----
    const size_t SZ_H1   = (size_t)N_NODES * 512;
    const size_t SZ_AL   = (size_t)N_NODES * 4;
    const size_t SZ_E    = (size_t)E_TOT * 4;
    const size_t SZ_OUT1 = (size_t)N_NODES * 512;

    float* ws   = (float*)d_ws;
    float* h1   = ws;                 ws += SZ_H1;
    float* as1  = ws;                 ws += SZ_AL;
    float* ad1  = ws;                 ws += SZ_AL;
    float* e1   = ws;                 ws += SZ_E;
    int*   m1   = (int*)ws;           ws += SZ_AL;
    float* den1 = ws;                 ws += SZ_AL;
    float* o1   = ws;                 ws += SZ_OUT1;
    // layer-2 aliases (h1 strip is free after layer-1 scatter)
    float* h2   = h1;                        // [50000,256]
    float* o2   = h1 + (size_t)N_NODES*256;  // [50000,256]

    const int T = 256;
    const int WPB = T / 32;
    const int MB = (N_NODES / 16 + 7) / 8;   // 128-row blocks along M (=391)

    // ========== Layer 1: H=4, C=128 ==========
    gemm_wmma_tiled<<<MB * (512 / BN), T, 0, stream>>>(xin, W1, h1, N_NODES, 512, 64);
    node_alpha<<<(N_NODES * 4 + WPB - 1) / WPB, T, 0, stream>>>(h1, aS1, aD1, as1, ad1, 4, 128);
    fill_i32<<<(N_NODES * 4 + T - 1) / T, T, 0, stream>>>(m1, NEG_KEY, N_NODES * 4);
    fill_f32<<<(N_NODES * 4 + T - 1) / T, T, 0, stream>>>(den1, 0.f, N_NODES * 4);
    fill_f32<<<((int)SZ_OUT1 + T - 1) / T, T, 0, stream>>>(o1, 0.f, (int)SZ_OUT1);
    edge_logits<<<(E_TOT + T - 1) / T, T, 0, stream>>>(ei, as1, ad1, e1, m1, 4);
    edge_expsum<<<(E_TOT + T - 1) / T, T, 0, stream>>>(ei, e1, m1, den1, 4);
    edge_scatter<<<(E_TOT + WPB - 1) / WPB, T, 0, stream>>>(ei, h1, e1, m1, den1, o1, 4, 128);
    bias_elu<<<((int)SZ_OUT1 + T - 1) / T, T, 0, stream>>>(o1, b1, (int)SZ_OUT1, 512);

    // ========== Layer 2: H=1, C=256 ==========
    gemm_wmma_tiled<<<MB * (256 / BN), T, 0, stream>>>(o1, W2, h2, N_NODES, 256, 512);
    node_alpha<<<(N_NODES + WPB - 1) / WPB, T, 0, stream>>>(h2, aS2, aD2, as1, ad1, 1, 256);
    fill_i32<<<(N_NODES + T - 1) / T, T, 0, stream>>>(m1, NEG_KEY, N_NODES);
    fill_f32<<<(N_NODES + T - 1) / T, T, 0, stream>>>(den1, 0.f, N_NODES);
    {
        int sz = N_NODES * 256;
        fill_f32<<<(sz + T - 1) / T, T, 0, stream>>>(o2, 0.f, sz);
    }
    edge_logits<<<(E_TOT + T - 1) / T, T, 0, stream>>>(ei, as1, ad1, e1, m1, 1);
    edge_expsum<<<(E_TOT + T - 1) / T, T, 0, stream>>>(ei, e1, m1, den1, 1);
    edge_scatter<<<(E_TOT + WPB - 1) / WPB, T, 0, stream>>>(ei, h2, e1, m1, den1, o2, 1, 256);

    // ========== mean over nodes + b2, broadcast to 8 rows ==========
    col_mean<<<256, 256, 0, stream>>>(o2, b2, out, 256, 8);

    (void)in_sizes; (void)n_in; (void)out_size; (void)ws_size;
}